// CustomerActor_15040975470999
// MI455X (gfx1250) — compile-verified
//
#include <hip/hip_runtime.h>
#include <hip/hip_bf16.h>

typedef __attribute__((ext_vector_type(2))) float v2f;
typedef __attribute__((ext_vector_type(8))) float v8f;

#define DIM 128
#define LDS_PITCH 132   // 128 + 4 pad: keeps float4 alignment, spreads LDS banks
#define BN_EPS 1e-5f
#define NLAYERS 3
#define MROWS 64        // rows per GEMM block (4 x 16-row WMMA tiles per wave)

// ---------------------------------------------------------------------------
// Edge scatter-add: one wave32 per edge, each lane moves 4 contiguous floats
// of the 128-float feature row. agg[dst] += h[src]. (agg pre-initialized = h,
// so after this kernel agg == h + segment_sum(h[src], dst) == z.)
// ---------------------------------------------------------------------------
__global__ __launch_bounds__(256)
void scatter_add_kernel(const float* __restrict__ h,
                        const int* __restrict__ src,
                        const int* __restrict__ dst,
                        float* __restrict__ agg, int E) {
    int gid  = blockIdx.x * blockDim.x + threadIdx.x;
    int e    = gid >> 5;
    int lane = gid & 31;
    if (e >= E) return;
    int s = src[e];
    int d = dst[e];
    const float4 v = *reinterpret_cast<const float4*>(h + (size_t)s * DIM + lane * 4);
    float* o = agg + (size_t)d * DIM + lane * 4;
    atomicAdd(o + 0, v.x);
    atomicAdd(o + 1, v.y);
    atomicAdd(o + 2, v.z);
    atomicAdd(o + 3, v.w);
}

// ---------------------------------------------------------------------------
// Fused GEMM: out[64 rows x 128 cols] = act( A @ W + bias )
// Block = 256 threads = 8 waves. Wave w owns the 16-col strip at w*16 and
// accumulates 4 stacked 16x16 tiles (M = row0 + 16*t), so each B fragment
// load feeds 4 V_WMMA_F32_16X16X4_F32 ops. A-tile (64x128) staged into LDS
// with CDNA5 async global->LDS copies (ASYNCcnt path).
// ---------------------------------------------------------------------------
__global__ __launch_bounds__(256)
void gin_mlp_gemm(const float* __restrict__ A,
                  const float* __restrict__ W,      // [DIM, DIM] row-major
                  const float* __restrict__ bias,   // [DIM]
                  float* __restrict__ out,
                  int do_relu, int N) {
    __shared__ float tile[MROWS * LDS_PITCH];

    const int row0 = blockIdx.x * MROWS;

    // Async stage 64x128 floats = 2048 b128 transfers, 8 per thread.
    for (int f = threadIdx.x; f < (MROWS * DIM / 4); f += 256) {
        int r  = f >> 5;               // 32 float4 per row
        int c4 = (f & 31) * 4;
        int rr = row0 + r; if (rr >= N) rr = N - 1;      // clamp (keeps EXEC full)
        unsigned lds_addr = (unsigned)(uintptr_t)&tile[r * LDS_PITCH + c4];
        unsigned goff     = (unsigned)(((size_t)rr * DIM + c4) * sizeof(float));
        asm volatile("global_load_async_to_lds_b128 %0, %1, %2 offset:0"
                     :: "v"(lds_addr), "v"(goff), "s"(A) : "memory");
    }
    asm volatile("s_wait_asynccnt 0" ::: "memory");
    __syncthreads();

    const int wave = threadIdx.x >> 5;
    const int lane = threadIdx.x & 31;
    const int half = lane >> 4;      // 0: lanes 0-15, 1: lanes 16-31
    const int m    = lane & 15;      // A row / B-C-D column within tile
    const int col  = wave * 16 + m;

    v8f acc0 = {}, acc1 = {}, acc2 = {}, acc3 = {};
    #pragma unroll
    for (int k0 = 0; k0 < DIM; k0 += 4) {
        const int k = k0 + 2 * half;
        v2f b;                        // B[K=k][N=col], B[K=k+1][N=col]
        b.x = W[(size_t)k * DIM + col];
        b.y = W[(size_t)(k + 1) * DIM + col];
        v2f a0, a1, a2, a3;           // A[M = m + 16*t][K=k..k+1]
        a0.x = tile[(m     ) * LDS_PITCH + k]; a0.y = tile[(m     ) * LDS_PITCH + k + 1];
        a1.x = tile[(m + 16) * LDS_PITCH + k]; a1.y = tile[(m + 16) * LDS_PITCH + k + 1];
        a2.x = tile[(m + 32) * LDS_PITCH + k]; a2.y = tile[(m + 32) * LDS_PITCH + k + 1];
        a3.x = tile[(m + 48) * LDS_PITCH + k]; a3.y = tile[(m + 48) * LDS_PITCH + k + 1];
        acc0 = __builtin_amdgcn_wmma_f32_16x16x4_f32(false, a0, false, b, (short)0, acc0, false, false);
        acc1 = __builtin_amdgcn_wmma_f32_16x16x4_f32(false, a1, false, b, (short)0, acc1, false, false);
        acc2 = __builtin_amdgcn_wmma_f32_16x16x4_f32(false, a2, false, b, (short)0, acc2, false, false);
        acc3 = __builtin_amdgcn_wmma_f32_16x16x4_f32(false, a3, false, b, (short)0, acc3, false, false);
    }

    const float bb = bias[col];
    v8f accs[4] = {acc0, acc1, acc2, acc3};
    #pragma unroll
    for (int t = 0; t < 4; ++t) {
        #pragma unroll
        for (int v = 0; v < 8; ++v) {
            int row = row0 + 16 * t + v + 8 * half;   // C/D: M = v + 8*(lane/16)
            if (row < N) {
                float val = accs[t][v] + bb;
                if (do_relu) val = fmaxf(val, 0.0f);
                out[(size_t)row * DIM + col] = val;
            }
        }
    }
}

// ---------------------------------------------------------------------------
// BatchNorm column statistics: thread t owns column t, grid-strides rows,
// then one atomicAdd per (block, column) into stats[0:128]=sum, [128:256]=sumsq.
// ---------------------------------------------------------------------------
__global__ __launch_bounds__(128)
void bn_stats_kernel(const float* __restrict__ z, float* __restrict__ stats, int N) {
    const int c = threadIdx.x;
    float s = 0.0f, s2 = 0.0f;
    for (int r = blockIdx.x; r < N; r += gridDim.x) {
        float v = z[(size_t)r * DIM + c];
        s  += v;
        s2 += v * v;
    }
    atomicAdd(&stats[c], s);
    atomicAdd(&stats[DIM + c], s2);
}

__global__ __launch_bounds__(128)
void bn_finalize_kernel(const float* __restrict__ stats,
                        const float* __restrict__ gamma,
                        const float* __restrict__ beta,
                        float* __restrict__ ss, float invN) {
    const int c  = threadIdx.x;
    float mu  = stats[c] * invN;
    float var = stats[DIM + c] * invN - mu * mu;   // biased variance
    float sc  = gamma[c] * rsqrtf(var + BN_EPS);
    ss[c]       = sc;
    ss[DIM + c] = beta[c] - mu * sc;
}

// h = relu(z * scale + shift), float4-vectorized.
__global__ __launch_bounds__(256)
void bn_apply_kernel(const float* __restrict__ z, const float* __restrict__ ss,
                     float* __restrict__ h, int n4) {
    int i = blockIdx.x * blockDim.x + threadIdx.x;
    if (i >= n4) return;
    int c = (i * 4) & (DIM - 1);
    float4 v = *reinterpret_cast<const float4*>(z + (size_t)i * 4);
    float4 o;
    o.x = fmaxf(v.x * ss[c + 0] + ss[DIM + c + 0], 0.0f);
    o.y = fmaxf(v.y * ss[c + 1] + ss[DIM + c + 1], 0.0f);
    o.z = fmaxf(v.z * ss[c + 2] + ss[DIM + c + 2], 0.0f);
    o.w = fmaxf(v.w * ss[c + 3] + ss[DIM + c + 3], 0.0f);
    *reinterpret_cast<float4*>(h + (size_t)i * 4) = o;
}

// ---------------------------------------------------------------------------
// Final FC: out[row] = dot(h[row], fc_w) + fc_b. One wave32 per row,
// lane loads float4, shfl tree-reduce within the wave.
// ---------------------------------------------------------------------------
__global__ __launch_bounds__(256)
void fc_kernel(const float* __restrict__ h, const float* __restrict__ w,
               const float* __restrict__ b, float* __restrict__ out, int N) {
    const int wave = threadIdx.x >> 5;
    const int lane = threadIdx.x & 31;
    const int row  = blockIdx.x * 8 + wave;
    if (row >= N) return;
    float4 hv = *reinterpret_cast<const float4*>(h + (size_t)row * DIM + lane * 4);
    float4 wv = *reinterpret_cast<const float4*>(w + lane * 4);
    float s = hv.x * wv.x + hv.y * wv.y + hv.z * wv.z + hv.w * wv.w;
    #pragma unroll
    for (int off = 16; off > 0; off >>= 1)
        s += __shfl_down(s, off, 32);
    if (lane == 0) out[row] = s + b[0];
}

// ---------------------------------------------------------------------------
extern "C" void kernel_launch(void* const* d_in, const int* in_sizes, int n_in,
                              void* d_out, int out_size, void* d_ws, size_t ws_size,
                              hipStream_t stream) {
    const float* x     = (const float*)d_in[0];
    const int*   ei    = (const int*)  d_in[1];   // [2, E]
    const float* W1    = (const float*)d_in[2];   // [L, D, D]
    const float* b1    = (const float*)d_in[3];   // [L, D]
    const float* W2    = (const float*)d_in[4];
    const float* b2    = (const float*)d_in[5];
    const float* gamma = (const float*)d_in[6];
    const float* beta  = (const float*)d_in[7];
    const float* fc_w  = (const float*)d_in[8];   // [D, 1]
    const float* fc_b  = (const float*)d_in[9];   // [1]

    const int N = in_sizes[0] / DIM;
    const int E = in_sizes[1] / 2;
    const size_t nd = (size_t)N * DIM;

    float* h     = (float*)d_ws;          // [N, D]
    float* t     = h + nd;                // [N, D] MLP hidden
    float* agg   = t + nd;                // [N, D] z = h + agg, reused as z2
    float* stats = agg + nd;              // [256]: col sums / sumsq
    float* ss    = stats + 256;           // [256]: scale / shift

    hipMemcpyAsync(h, x, nd * sizeof(float), hipMemcpyDeviceToDevice, stream);

    const int gemm_blocks    = (N + MROWS - 1) / MROWS;
    const int scatter_blocks = (E * 32 + 255) / 256;
    const int apply_blocks   = ((int)(nd / 4) + 255) / 256;

    for (int l = 0; l < NLAYERS; ++l) {
        // agg := h, then agg += segment_sum(h[src], dst)  =>  agg == z
        hipMemcpyAsync(agg, h, nd * sizeof(float), hipMemcpyDeviceToDevice, stream);
        scatter_add_kernel<<<scatter_blocks, 256, 0, stream>>>(h, ei, ei + E, agg, E);

        // t  = relu(z @ W1[l] + b1[l])
        gin_mlp_gemm<<<gemm_blocks, 256, 0, stream>>>(
            agg, W1 + (size_t)l * DIM * DIM, b1 + (size_t)l * DIM, t, 1, N);
        // z2 = t @ W2[l] + b2[l]   (reuse agg buffer)
        gin_mlp_gemm<<<gemm_blocks, 256, 0, stream>>>(
            t, W2 + (size_t)l * DIM * DIM, b2 + (size_t)l * DIM, agg, 0, N);

        hipMemsetAsync(stats, 0, 256 * sizeof(float), stream);
        bn_stats_kernel<<<512, 128, 0, stream>>>(agg, stats, N);
        bn_finalize_kernel<<<1, 128, 0, stream>>>(
            stats, gamma + (size_t)l * DIM, beta + (size_t)l * DIM, ss, 1.0f / (float)N);
        bn_apply_kernel<<<apply_blocks, 256, 0, stream>>>(agg, ss, h, (int)(nd / 4));
    }

    fc_kernel<<<(N + 7) / 8, 256, 0, stream>>>(h, fc_w, fc_b, (float*)d_out, N);
}